// LorentzKG_62380105008051
// MI455X (gfx1250) — compile-verified
//
#include <hip/hip_runtime.h>
#include <cmath>
#include <cstdint>

// ---------------------------------------------------------------------------
// LorentzKG scoring, MI455X (gfx1250, wave32).
//   Kernel 1: per-relation precompute (kills ~35 transcendentals/triple).
//   Kernel 2: per-triple gather+score; relation table staged to LDS via the
//             CDNA5 Tensor Data Mover (tensor_load_to_lds + s_wait_tensorcnt).
// ---------------------------------------------------------------------------

namespace {
constexpr int kNR   = 1000;
constexpr int kD    = 32;
constexpr int kHalf = 16;
constexpr int kB    = 1 << 20;

// Record A (staged to LDS): c[16], s[16], cosh(rap0), sinh(rap0), cosh(vn), sinh(vn)/vn
constexpr int kRecAF = 36;                       // floats (144 B, 16B aligned)
// Record B (stays in L2):  vs[32] = 0.1 * translation
constexpr int kRecBF = 32;                       // floats (128 B)
constexpr int kTabABytes = kNR * kRecAF * 4;     // 144000 B of LDS per workgroup
}

typedef float        f4    __attribute__((ext_vector_type(4), aligned(4)));
typedef unsigned int u32x4 __attribute__((ext_vector_type(4)));
typedef int          i32x4 __attribute__((ext_vector_type(4)));
typedef int          i32x8 __attribute__((ext_vector_type(8)));

// ---------------------------------------------------------------------------
// Kernel 1: derive per-relation constants. 1000 relations -> trivial cost.
// ---------------------------------------------------------------------------
__global__ void lkg_prep_kernel(const float* __restrict__ rel_rot,
                                const float* __restrict__ rel_boost,
                                const float* __restrict__ rel_trans,
                                float* __restrict__ recA,
                                float* __restrict__ recB) {
    const int r = blockIdx.x * blockDim.x + threadIdx.x;
    if (r >= kNR) return;

    const float* rot = rel_rot + (size_t)r * kD;
    float* A = recA + (size_t)r * kRecAF;
#pragma unroll
    for (int i = 0; i < kHalf; ++i) {
        const float th = rot[i];
        A[i]         = cosf(th);
        A[kHalf + i] = sinf(th);
    }

    float rap = rel_boost[(size_t)r * kD];           // only component 0 is used
    rap = fminf(2.0f, fmaxf(-2.0f, rap));
    A[32] = coshf(rap);
    A[33] = sinhf(rap);

    const float* tr = rel_trans + (size_t)r * kD;
    float* Bv = recB + (size_t)r * kRecBF;
    float ss = 0.0f;
#pragma unroll
    for (int j = 0; j < kD; ++j) {
        const float v = 0.1f * tr[j];
        Bv[j] = v;
        ss = fmaf(v, v, ss);
    }
    const float vn = sqrtf(fmaxf(ss, 1e-6f));
    A[34] = coshf(vn);
    A[35] = sinhf(vn) / vn;
}

// ---------------------------------------------------------------------------
// Kernel 2: main scoring kernel. One triple per lane, grid-stride.
//   - Record-A table (144 KB) staged to LDS with one TDM descriptor per WG.
//   - Entity rows loaded as dword-aligned b128 vectors.
// ---------------------------------------------------------------------------
__global__ void __launch_bounds__(1024)
lkg_main_kernel(const int* __restrict__ heads,
                const int* __restrict__ rels,
                const int* __restrict__ tails,
                const float* __restrict__ emb,      // (NE, 33)
                const float* __restrict__ bias,     // (NE,)
                const float* __restrict__ recB,     // (NR, 32) vs table
                float* __restrict__ out,
                u32x4 g0, i32x8 g1, i32x4 g2, i32x4 g3) {
    extern __shared__ float s_recA[];               // kTabABytes

    // --- CDNA5 TDM: async copy recA table (global -> LDS), one op per WG ---
    if (threadIdx.x == 0) {
        g0.y = (unsigned int)(uintptr_t)(void*)s_recA;   // lds_addr (low 32 bits)
        const i32x8 g4 = { 0, 0, 0, 0, 0, 0, 0, 0 };     // zero-filled trailing group
        __builtin_amdgcn_tensor_load_to_lds(g0, g1, g2, g3, g4, 0);
    }
    __builtin_amdgcn_s_wait_tensorcnt(0);           // waves w/ TENSORcnt==0 fall through
    __syncthreads();

    const int stride = gridDim.x * blockDim.x;
    for (int i = blockIdx.x * blockDim.x + threadIdx.x; i < kB; i += stride) {
        const int hi = heads[i];
        const int ri = rels[i];
        const int ti = tails[i];

        const float* hrow = emb + (size_t)hi * (kD + 1);
        const float* trow = emb + (size_t)ti * (kD + 1);
        __builtin_prefetch(trow, 0, 3);             // global_prefetch_b8 for the tail row
        const float bsum = bias[hi] + bias[ti];

        const float* A = s_recA + ri * kRecAF;      // LDS, 16B-aligned (ri*144)

        // head row: x0 + 32 spatial (dword-aligned b128 loads)
        const float x0 = hrow[0];
        const f4* hv = (const f4*)(hrow + 1);
        f4 sp[8];
#pragma unroll
        for (int k = 0; k < 8; ++k) sp[k] = hv[k];

        // rotation: ns[0:16) = c*a - s*b ; ns[16:32) = s*a + c*b
        const f4* cv = (const f4*)A;
        const f4* sv = (const f4*)(A + kHalf);
        float ns[kD];
#pragma unroll
        for (int k = 0; k < 4; ++k) {
            const f4 c = cv[k], s = sv[k], av = sp[k], bv = sp[4 + k];
#pragma unroll
            for (int e = 0; e < 4; ++e) {
                ns[4 * k + e]      = c[e] * av[e] - s[e] * bv[e];
                ns[16 + 4 * k + e] = fmaf(s[e], av[e], c[e] * bv[e]);
            }
        }

        // boost on component 1 only (h_boost time component is never consumed)
        const float c0 = A[32], s0 = A[33], ch = A[34], shvn = A[35];
        ns[0] = fmaf(x0, s0, ns[0] * c0);

        // translation: rs = ch*ns + shvn*vs ; accumulate sum of squares
        const f4* vv = (const f4*)(recB + (size_t)ri * kRecBF);
        float ss2 = 0.0f;
#pragma unroll
        for (int k = 0; k < 8; ++k) {
            const f4 v = vv[k];
#pragma unroll
            for (int e = 0; e < 4; ++e) {
                const float rj = fmaf(ch, ns[4 * k + e], shvn * v[e]);
                ns[4 * k + e] = rj;
                ss2 = fmaf(rj, rj, ss2);
            }
        }
        const float x0t = sqrtf(1.0f + ss2);        // project(res[1:]) time component

        // Lorentz inner with tail
        const float t0 = trow[0];
        const f4* tv = (const f4*)(trow + 1);
        float acc = 0.0f;
#pragma unroll
        for (int k = 0; k < 8; ++k) {
            const f4 t4 = tv[k];
#pragma unroll
            for (int e = 0; e < 4; ++e) acc = fmaf(ns[4 * k + e], t4[e], acc);
        }
        const float inner = acc - x0t * t0;
        const float ic    = fmaxf(-inner, 1.0f + 1e-6f);
        const float dist  = acoshf(ic);
        out[i] = fmaf(-dist, dist, bsum);
    }
}

// ---------------------------------------------------------------------------
// Host launcher. Input order: heads, relations, tails, entity_emb,
// rel_boost_w, rel_rot_w, rel_trans_w, ent_bias_w.
// ---------------------------------------------------------------------------
extern "C" void kernel_launch(void* const* d_in, const int* in_sizes, int n_in,
                              void* d_out, int out_size, void* d_ws, size_t ws_size,
                              hipStream_t stream) {
    (void)in_sizes; (void)n_in; (void)out_size; (void)ws_size;

    const int*   heads     = (const int*)d_in[0];
    const int*   rels      = (const int*)d_in[1];
    const int*   tails     = (const int*)d_in[2];
    const float* emb       = (const float*)d_in[3];
    const float* rel_boost = (const float*)d_in[4];
    const float* rel_rot   = (const float*)d_in[5];
    const float* rel_trans = (const float*)d_in[6];
    const float* bias      = (const float*)d_in[7];
    float*       out       = (float*)d_out;

    float* recA = (float*)d_ws;                              // 144000 B
    float* recB = recA + (size_t)kNR * kRecAF;               // +128000 B

    lkg_prep_kernel<<<(kNR + 255) / 256, 256, 0, stream>>>(
        rel_rot, rel_boost, rel_trans, recA, recB);

    // --- Build TDM D# (groups per CDNA5 ISA): 2-D tile,
    //     data_size=8B, rows of 18 qwords (144 B) x 1000 rows, contiguous. ---
    const uint64_t ga     = (uint64_t)(uintptr_t)recA;
    const uint32_t rowQW  = (kRecAF * 4) / 8;                // 18
    const uint32_t rows   = kNR;                             // 1000
    const uint64_t d1strd = (uint64_t)rowQW * rows;          // 18000

    u32x4 g0 = { 1u,                                         // count=1 valid descriptor
                 0u,                                         // lds_addr (patched in kernel)
                 (uint32_t)ga,                               // global_addr[31:0]
                 ((uint32_t)((ga >> 32) & 0x1FFFFFFu)) | 0x80000000u }; // addr[56:32] | type=2
    i32x8 g1;
    g1[0] = 0x00030000;                                      // wg_mask=0, data_size=3 (8B)
    g1[1] = (int)(rowQW << 16);                              // tensor_dim0[15:0] @ bits79:48
    g1[2] = (int)(rows  << 16);                              // tensor_dim1[15:0] @ bits111:80
    g1[3] = (int)(rowQW << 16);                              // tile_dim0 @ bits127:112
    g1[4] = (int)rows;                                       // tile_dim1; tile_dim2=0
    g1[5] = (int)rowQW;                                      // tensor_dim0_stride low32
    g1[6] = (int)((d1strd & 0xFFFFull) << 16);               // dim0_stride hi=0 | dim1_stride lo16
    g1[7] = (int)(d1strd >> 16);                             // dim1_stride hi
    i32x4 g2 = { 1, 1, 0, 0 };                               // tensor_dim2=1, tensor_dim3=1 (unused)
    i32x4 g3 = { 0, 0, 0, 0 };

    (void)hipFuncSetAttribute((const void*)lkg_main_kernel,
                              hipFuncAttributeMaxDynamicSharedMemorySize, kTabABytes);

    lkg_main_kernel<<<512, 1024, kTabABytes, stream>>>(
        heads, rels, tails, emb, bias, recB, out, g0, g1, g2, g3);
}